// NormalizationLayer_25099788878674
// MI455X (gfx1250) — compile-verified
//
#include <hip/hip_runtime.h>
#include <cstdint>

// Problem constants from the reference: B=4096, N=10000, K=100
#define BATCHES 4096
#define NNODES  10000
#define KCAND   100
#define TPB     256                      // 8 wave32 waves per block
#define NWAVES  (TPB / 32)
#define V4_PER_BATCH ((NNODES * 2) / 4)  // 5000 float4 per batch (80 KB)

typedef float v4f __attribute__((ext_vector_type(4)));

__global__ __launch_bounds__(TPB) void bbox_normalize_kernel(
    const float* __restrict__ nodes,   // [B, N, 2] f32
    const int*   __restrict__ cand,    // [B, K] i32 (JAX x64-off => int32)
    float*       __restrict__ out)     // [B, N, 2] f32
{
    const int b    = blockIdx.x;
    const int tid  = threadIdx.x;
    const int lane = tid & 31;
    const int wave = tid >> 5;

    __shared__ float2 s_pts[KCAND];
    __shared__ float  s_xmn[NWAVES], s_xmx[NWAVES], s_ymn[NWAVES], s_ymx[NWAVES];
    __shared__ int    s_any[NWAVES];
    __shared__ float  s_px, s_py, s_r;
    __shared__ int    s_anyv;

    // ---- Phase 1: async-gather candidate points (8 B each) into LDS ----
    int myvalid = 0;
    if (tid < KCAND) {
        const int idx  = cand[(size_t)b * KCAND + tid];
        myvalid        = (idx != -1);
        const int safe = myvalid ? idx : 0;
        const float* gp = nodes + ((size_t)b * NNODES + (size_t)safe) * 2;
        unsigned           lds_dst = (unsigned)(uintptr_t)(void*)(&s_pts[tid]);
        unsigned long long gaddr   = (unsigned long long)(uintptr_t)gp;
        // CDNA5 async global -> LDS copy, tracked by ASYNCcnt
        asm volatile("global_load_async_to_lds_b64 %0, %1, off"
                     :: "v"(lds_dst), "v"(gaddr)
                     : "memory");
    }
    asm volatile("s_wait_asynccnt 0" ::: "memory");
    __syncthreads();

    // ---- Phase 2: per-thread partials, then wave32 + LDS reduction ----
    float xmn =  __builtin_inff(), xmx = -__builtin_inff();
    float ymn =  __builtin_inff(), ymx = -__builtin_inff();
    if (tid < KCAND && myvalid) {
        const float2 p = s_pts[tid];
        xmn = xmx = p.x;
        ymn = ymx = p.y;
    }
    #pragma unroll
    for (int off = 16; off > 0; off >>= 1) {   // wave32: 5 butterfly steps
        xmn = fminf(xmn, __shfl_xor(xmn, off, 32));
        xmx = fmaxf(xmx, __shfl_xor(xmx, off, 32));
        ymn = fminf(ymn, __shfl_xor(ymn, off, 32));
        ymx = fmaxf(ymx, __shfl_xor(ymx, off, 32));
        myvalid |= __shfl_xor(myvalid, off, 32);
    }
    if (lane == 0) {
        s_xmn[wave] = xmn; s_xmx[wave] = xmx;
        s_ymn[wave] = ymn; s_ymx[wave] = ymx;
        s_any[wave] = myvalid;
    }
    __syncthreads();
    if (tid == 0) {
        float axmn =  __builtin_inff(), axmx = -__builtin_inff();
        float aymn =  __builtin_inff(), aymx = -__builtin_inff();
        int anyv = 0;
        #pragma unroll
        for (int w = 0; w < NWAVES; ++w) {
            axmn = fminf(axmn, s_xmn[w]); axmx = fmaxf(axmx, s_xmx[w]);
            aymn = fminf(aymn, s_ymn[w]); aymx = fmaxf(aymx, s_ymx[w]);
            anyv |= s_any[w];
        }
        float span = fmaxf(axmx - axmn, aymx - aymn);
        span       = fmaxf(span, 1e-6f);           // jnp.clip(span, 1e-6)
        s_px   = axmn;
        s_py   = aymn;
        s_r    = 1.0f / span;
        s_anyv = anyv;
    }
    __syncthreads();

    // ---- Phase 3: stream 80 KB/batch through clip(r*(v-min),0,1) ----
    const float xm = s_px, ym = s_py, r = s_r;
    const int anyv = s_anyv;
    const v4f* __restrict__ src = (const v4f*)(nodes + (size_t)b * NNODES * 2);
    v4f*       __restrict__ dst = (v4f*)(out  + (size_t)b * NNODES * 2);

    if (anyv) {
        for (int i = tid; i < V4_PER_BATCH; i += TPB) {
            v4f v = __builtin_nontemporal_load(src + i);   // (x,y,x,y)
            v4f w;
            w.x = fminf(fmaxf((v.x - xm) * r, 0.0f), 1.0f);
            w.y = fminf(fmaxf((v.y - ym) * r, 0.0f), 1.0f);
            w.z = fminf(fmaxf((v.z - xm) * r, 0.0f), 1.0f);
            w.w = fminf(fmaxf((v.w - ym) * r, 0.0f), 1.0f);
            __builtin_nontemporal_store(w, dst + i);
        }
    } else {
        for (int i = tid; i < V4_PER_BATCH; i += TPB) {
            __builtin_nontemporal_store(__builtin_nontemporal_load(src + i), dst + i);
        }
    }
}

extern "C" void kernel_launch(void* const* d_in, const int* in_sizes, int n_in,
                              void* d_out, int out_size, void* d_ws, size_t ws_size,
                              hipStream_t stream) {
    const float* nodes = (const float*)d_in[0];
    const int*   cand  = (const int*)d_in[1];
    float*       out   = (float*)d_out;
    (void)in_sizes; (void)n_in; (void)out_size; (void)d_ws; (void)ws_size;
    hipLaunchKernelGGL(bbox_normalize_kernel, dim3(BATCHES), dim3(TPB), 0, stream,
                       nodes, cand, out);
}